// MoEPositionwiseFeedForward_54520314855563
// MI455X (gfx1250) — compile-verified
//
#include <hip/hip_runtime.h>

typedef float v2f __attribute__((ext_vector_type(2)));
typedef float v8f __attribute__((ext_vector_type(8)));

namespace {
constexpr int T_TOK = 8192;   // B*S
constexpr int DIM   = 2048;   // D
constexpr int FDIM  = 1024;   // F
constexpr int NEXP  = 7;      // routed experts
constexpr int CAP   = 1170;   // int(T/NEXP)

constexpr int BM  = 64;
constexpr int BN  = 128;
constexpr int BK  = 16;
constexpr int LDK = BK + 2;   // 18: even -> 8B-aligned (k,k+1) pairs, conflict-free b64 reads
}

// ---------------- init scratch ----------------
__global__ void moe_init_kernel(int* __restrict__ tok_slot,
                                int* __restrict__ slot_tok,
                                float* __restrict__ slot_w) {
    int i = blockIdx.x * 256 + threadIdx.x;
    if (i < T_TOK * 2) tok_slot[i] = -1;
    if (i < NEXP * CAP) { slot_tok[i] = 0; slot_w[i] = 0.0f; }
}

// ---------------- gating: logits -> sigmoid -> top2 ----------------
__global__ __launch_bounds__(256) void moe_gate_kernel(
    const float* __restrict__ x, const float* __restrict__ gw,
    const float* __restrict__ bias, float* __restrict__ aff,
    int* __restrict__ top2) {
    __shared__ float red[NEXP][272];
    const int t = blockIdx.x;
    const int tid = threadIdx.x;
    float acc[NEXP];
#pragma unroll
    for (int e = 0; e < NEXP; ++e) acc[e] = 0.0f;
    const float* xr = x + (size_t)t * DIM;
    for (int d = tid; d < DIM; d += 256) {
        const float xv = xr[d];
        const float* g = gw + (size_t)d * NEXP;
#pragma unroll
        for (int e = 0; e < NEXP; ++e) acc[e] += xv * g[e];
    }
#pragma unroll
    for (int e = 0; e < NEXP; ++e) red[e][tid] = acc[e];
    __syncthreads();
    for (int s = 128; s > 0; s >>= 1) {
        if (tid < s) {
#pragma unroll
            for (int e = 0; e < NEXP; ++e) red[e][tid] += red[e][tid + s];
        }
        __syncthreads();
    }
    if (tid == 0) {
        float a[NEXP];
#pragma unroll
        for (int e = 0; e < NEXP; ++e) {
            const float logit = red[e][0] + bias[e];
            a[e] = 1.0f / (1.0f + expf(-logit));
            aff[(size_t)t * NEXP + e] = a[e];
        }
        int b0 = 0;
#pragma unroll
        for (int e = 1; e < NEXP; ++e) if (a[e] > a[b0]) b0 = e;      // stable argmax
        int b1 = -1;
#pragma unroll
        for (int e = 0; e < NEXP; ++e) {
            if (e == b0) continue;
            if (b1 < 0 || a[e] > a[b1]) b1 = e;                       // stable 2nd
        }
        top2[t * 2 + 0] = b0;
        top2[t * 2 + 1] = b1;
    }
}

// -------- per-expert capacity ranking (stable, matches lax.top_k) --------
__global__ __launch_bounds__(256) void moe_select_kernel(
    const float* __restrict__ aff, const int* __restrict__ top2,
    int* __restrict__ slot_tok, float* __restrict__ slot_w,
    int* __restrict__ tok_slot) {
    __shared__ float sc[256];
    const int e = blockIdx.y;
    const int t = blockIdx.x * 256 + threadIdx.x;
    const int m0 = top2[t * 2 + 0];
    const int m1 = top2[t * 2 + 1];
    const bool mine = (m0 == e) || (m1 == e);
    const float my = mine ? aff[(size_t)t * NEXP + e] : -1.0f;
    int count = 0;
    for (int base = 0; base < T_TOK; base += 256) {
        const int tt = base + threadIdx.x;
        const int o0 = top2[tt * 2 + 0];
        const int o1 = top2[tt * 2 + 1];
        sc[threadIdx.x] = (o0 == e || o1 == e) ? aff[(size_t)tt * NEXP + e] : -1.0f;
        __syncthreads();
        if (mine) {
            for (int i = 0; i < 256; ++i) {
                const float s = sc[i];
                const int tt2 = base + i;
                count += ((s > my) || (s == my && tt2 < t)) ? 1 : 0;
            }
        }
        __syncthreads();
    }
    if (mine && count < CAP) {
        const int slot = e * CAP + count;           // ranks are unique -> no atomics
        slot_tok[slot] = t;
        slot_w[slot] = my;
        const int which = (m0 == e) ? 0 : 1;
        tok_slot[t * 2 + which] = slot;
    }
}

// ---------------- GEMM1 fused with SwiGLU: G = x1 * silu(x2) ----------------
__global__ __launch_bounds__(256) void moe_gemm1_swiglu_kernel(
    const float* __restrict__ X, const int* __restrict__ rowmap,
    const float* __restrict__ W1base, long w1_stride,
    float* __restrict__ Gbase, long g_stride, int Mvalid) {
    __shared__ float As[BM][LDK];    // A tile, [m][k]
    __shared__ float B1t[BN][LDK];   // B tiles TRANSPOSED: [n][k] -> (k,k+1) contiguous
    __shared__ float B2t[BN][LDK];

    const int e = blockIdx.z;
    const float* W = W1base + (long)e * w1_stride;   // [DIM, 2*FDIM]
    float* Gout = Gbase + (long)e * g_stride;        // [Mvalid, FDIM]
    const int* rmap = rowmap ? (rowmap + (long)e * Mvalid) : nullptr;

    const int row0 = blockIdx.y * BM;
    const int col0 = blockIdx.x * BN;     // within FDIM
    const int tid = threadIdx.x;
    const int lane = tid & 31;
    const int wid = tid >> 5;
    const int wrow = (wid >> 2) * 32;     // 0 / 32
    const int wcol = (wid & 3) * 32;      // 0..96
    const int l15 = lane & 15;
    const int lhi = lane >> 4;

    const int ar = tid >> 2;              // 0..63
    const int acq = (tid & 3) * 4;        // 0,4,8,12
    int arow = row0 + ar;
    if (arow > Mvalid - 1) arow = Mvalid - 1;
    const int atok = rmap ? rmap[arow] : arow;
    const float* Arow = X + (size_t)atok * DIM;

    const int bkr = tid >> 5;             // 0..7
    const int bcq = (tid & 31) * 4;       // 0..124

    const v8f vz = {0.f, 0.f, 0.f, 0.f, 0.f, 0.f, 0.f, 0.f};
    v8f acc1[2][2] = {{vz, vz}, {vz, vz}};
    v8f acc2[2][2] = {{vz, vz}, {vz, vz}};

    for (int k0 = 0; k0 < DIM; k0 += BK) {
        // stage A (64x16), row-major
        const float4 av = *(const float4*)(Arow + k0 + acq);
        As[ar][acq + 0] = av.x; As[ar][acq + 1] = av.y;
        As[ar][acq + 2] = av.z; As[ar][acq + 3] = av.w;
        // stage B1 (x1 cols) and B2 (x2 cols), transposed into [n][k]
        const float* w0 = W + (size_t)(k0 + bkr) * (2 * FDIM) + col0 + bcq;
        const float* w8 = W + (size_t)(k0 + bkr + 8) * (2 * FDIM) + col0 + bcq;
        const float4 b1a = *(const float4*)(w0);
        const float4 b1b = *(const float4*)(w8);
        const float4 b2a = *(const float4*)(w0 + FDIM);
        const float4 b2b = *(const float4*)(w8 + FDIM);
        B1t[bcq + 0][bkr] = b1a.x; B1t[bcq + 1][bkr] = b1a.y;
        B1t[bcq + 2][bkr] = b1a.z; B1t[bcq + 3][bkr] = b1a.w;
        B1t[bcq + 0][bkr + 8] = b1b.x; B1t[bcq + 1][bkr + 8] = b1b.y;
        B1t[bcq + 2][bkr + 8] = b1b.z; B1t[bcq + 3][bkr + 8] = b1b.w;
        B2t[bcq + 0][bkr] = b2a.x; B2t[bcq + 1][bkr] = b2a.y;
        B2t[bcq + 2][bkr] = b2a.z; B2t[bcq + 3][bkr] = b2a.w;
        B2t[bcq + 0][bkr + 8] = b2b.x; B2t[bcq + 1][bkr + 8] = b2b.y;
        B2t[bcq + 2][bkr + 8] = b2b.z; B2t[bcq + 3][bkr + 8] = b2b.w;
        __syncthreads();
#pragma unroll
        for (int kk = 0; kk < BK; kk += 4) {
            const int kb = kk + lhi * 2;
            v2f a0, a1, b0, b1, c0, c1;
            a0.x = As[wrow + l15][kb];           a0.y = As[wrow + l15][kb + 1];
            a1.x = As[wrow + 16 + l15][kb];      a1.y = As[wrow + 16 + l15][kb + 1];
            b0.x = B1t[wcol + l15][kb];          b0.y = B1t[wcol + l15][kb + 1];
            b1.x = B1t[wcol + 16 + l15][kb];     b1.y = B1t[wcol + 16 + l15][kb + 1];
            c0.x = B2t[wcol + l15][kb];          c0.y = B2t[wcol + l15][kb + 1];
            c1.x = B2t[wcol + 16 + l15][kb];     c1.y = B2t[wcol + 16 + l15][kb + 1];
            acc1[0][0] = __builtin_amdgcn_wmma_f32_16x16x4_f32(false, a0, false, b0, (short)0, acc1[0][0], false, false);
            acc1[0][1] = __builtin_amdgcn_wmma_f32_16x16x4_f32(false, a0, false, b1, (short)0, acc1[0][1], false, false);
            acc1[1][0] = __builtin_amdgcn_wmma_f32_16x16x4_f32(false, a1, false, b0, (short)0, acc1[1][0], false, false);
            acc1[1][1] = __builtin_amdgcn_wmma_f32_16x16x4_f32(false, a1, false, b1, (short)0, acc1[1][1], false, false);
            acc2[0][0] = __builtin_amdgcn_wmma_f32_16x16x4_f32(false, a0, false, c0, (short)0, acc2[0][0], false, false);
            acc2[0][1] = __builtin_amdgcn_wmma_f32_16x16x4_f32(false, a0, false, c1, (short)0, acc2[0][1], false, false);
            acc2[1][0] = __builtin_amdgcn_wmma_f32_16x16x4_f32(false, a1, false, c0, (short)0, acc2[1][0], false, false);
            acc2[1][1] = __builtin_amdgcn_wmma_f32_16x16x4_f32(false, a1, false, c1, (short)0, acc2[1][1], false, false);
        }
        __syncthreads();
    }
#pragma unroll
    for (int i = 0; i < 2; ++i)
#pragma unroll
        for (int j = 0; j < 2; ++j)
#pragma unroll
            for (int v = 0; v < 8; ++v) {
                const int grow = row0 + wrow + i * 16 + lhi * 8 + v;
                if (grow < Mvalid) {
                    const int gcol = col0 + wcol + j * 16 + l15;
                    const float x1 = acc1[i][j][v];
                    const float x2 = acc2[i][j][v];
                    const float s = 1.0f / (1.0f + __expf(-x2));
                    Gout[(size_t)grow * FDIM + gcol] = x1 * x2 * s;
                }
            }
}

// ---------------- GEMM2: Y = G @ W2 ----------------
__global__ __launch_bounds__(256) void moe_gemm2_kernel(
    const float* __restrict__ Gbase, long g_stride,
    const float* __restrict__ W2base, long w2_stride,
    float* __restrict__ Ybase, long y_stride, int Mvalid) {
    __shared__ float As[BM][LDK];
    __shared__ float Bt[BN][LDK];    // transposed [n][k]

    const int e = blockIdx.z;
    const float* G = Gbase + (long)e * g_stride;     // [Mvalid, FDIM]
    const float* W = W2base + (long)e * w2_stride;   // [FDIM, DIM]
    float* Y = Ybase + (long)e * y_stride;           // [Mvalid, DIM]

    const int row0 = blockIdx.y * BM;
    const int col0 = blockIdx.x * BN;   // within DIM
    const int tid = threadIdx.x;
    const int lane = tid & 31;
    const int wid = tid >> 5;
    const int wrow = (wid >> 2) * 32;
    const int wcol = (wid & 3) * 32;
    const int l15 = lane & 15;
    const int lhi = lane >> 4;

    const int ar = tid >> 2;
    const int acq = (tid & 3) * 4;
    int arow = row0 + ar;
    if (arow > Mvalid - 1) arow = Mvalid - 1;
    const float* Arow = G + (size_t)arow * FDIM;

    const int bkr = tid >> 5;
    const int bcq = (tid & 31) * 4;

    const v8f vz = {0.f, 0.f, 0.f, 0.f, 0.f, 0.f, 0.f, 0.f};
    v8f acc[2][2] = {{vz, vz}, {vz, vz}};

    for (int k0 = 0; k0 < FDIM; k0 += BK) {
        const float4 av = *(const float4*)(Arow + k0 + acq);
        As[ar][acq + 0] = av.x; As[ar][acq + 1] = av.y;
        As[ar][acq + 2] = av.z; As[ar][acq + 3] = av.w;
        const float4 ba = *(const float4*)(W + (size_t)(k0 + bkr) * DIM + col0 + bcq);
        const float4 bb = *(const float4*)(W + (size_t)(k0 + bkr + 8) * DIM + col0 + bcq);
        Bt[bcq + 0][bkr] = ba.x; Bt[bcq + 1][bkr] = ba.y;
        Bt[bcq + 2][bkr] = ba.z; Bt[bcq + 3][bkr] = ba.w;
        Bt[bcq + 0][bkr + 8] = bb.x; Bt[bcq + 1][bkr + 8] = bb.y;
        Bt[bcq + 2][bkr + 8] = bb.z; Bt[bcq + 3][bkr + 8] = bb.w;
        __syncthreads();
#pragma unroll
        for (int kk = 0; kk < BK; kk += 4) {
            const int kb = kk + lhi * 2;
            v2f a0, a1, b0, b1;
            a0.x = As[wrow + l15][kb];           a0.y = As[wrow + l15][kb + 1];
            a1.x = As[wrow + 16 + l15][kb];      a1.y = As[wrow + 16 + l15][kb + 1];
            b0.x = Bt[wcol + l15][kb];           b0.y = Bt[wcol + l15][kb + 1];
            b1.x = Bt[wcol + 16 + l15][kb];      b1.y = Bt[wcol + 16 + l15][kb + 1];
            acc[0][0] = __builtin_amdgcn_wmma_f32_16x16x4_f32(false, a0, false, b0, (short)0, acc[0][0], false, false);
            acc[0][1] = __builtin_amdgcn_wmma_f32_16x16x4_f32(false, a0, false, b1, (short)0, acc[0][1], false, false);
            acc[1][0] = __builtin_amdgcn_wmma_f32_16x16x4_f32(false, a1, false, b0, (short)0, acc[1][0], false, false);
            acc[1][1] = __builtin_amdgcn_wmma_f32_16x16x4_f32(false, a1, false, b1, (short)0, acc[1][1], false, false);
        }
        __syncthreads();
    }
#pragma unroll
    for (int i = 0; i < 2; ++i)
#pragma unroll
        for (int j = 0; j < 2; ++j)
#pragma unroll
            for (int v = 0; v < 8; ++v) {
                const int grow = row0 + wrow + i * 16 + lhi * 8 + v;
                if (grow < Mvalid) {
                    const int gcol = col0 + wcol + j * 16 + l15;
                    Y[(size_t)grow * DIM + gcol] = acc[i][j][v];
                }
            }
}

// -------- final gather: out[t] = shared_Y[t] + sum_k w * Yr[slot_k] --------
__global__ __launch_bounds__(256) void moe_gather_kernel(
    float* __restrict__ out, const float* __restrict__ Yr,
    const int* __restrict__ tok_slot, const float* __restrict__ slot_w) {
    const size_t i = (size_t)blockIdx.x * 256 + threadIdx.x;  // T*DIM/4 threads
    const int t = (int)(i / (DIM / 4));
    const int dq = (int)(i % (DIM / 4)) * 4;
    float4 v = *(float4*)(out + (size_t)t * DIM + dq);
#pragma unroll
    for (int k = 0; k < 2; ++k) {
        const int s = tok_slot[t * 2 + k];
        if (s >= 0) {
            const float w = slot_w[s];
            const float4 y = *(const float4*)(Yr + (size_t)s * DIM + dq);
            v.x += w * y.x; v.y += w * y.y; v.z += w * y.z; v.w += w * y.w;
        }
    }
    *(float4*)(out + (size_t)t * DIM + dq) = v;
}

extern "C" void kernel_launch(void* const* d_in, const int* in_sizes, int n_in,
                              void* d_out, int out_size, void* d_ws, size_t ws_size,
                              hipStream_t stream) {
    const float* x       = (const float*)d_in[0];
    const float* gate_w  = (const float*)d_in[1];
    const float* bias    = (const float*)d_in[2];
    const float* sw1     = (const float*)d_in[3];   // [1, D, 2F]
    const float* sw2     = (const float*)d_in[4];   // [1, F, D]
    const float* rw1     = (const float*)d_in[5];   // [E, D, 2F]
    const float* rw2     = (const float*)d_in[6];   // [E, F, D]
    float* out = (float*)d_out;

    char* ws = (char*)d_ws;
    size_t off = 0;
    auto carve = [&](size_t bytes) -> void* {
        void* p = ws + off;
        off = (off + bytes + 255) & ~(size_t)255;
        return p;
    };
    float* aff      = (float*)carve((size_t)T_TOK * NEXP * sizeof(float));
    int*   top2     = (int*)  carve((size_t)T_TOK * 2 * sizeof(int));
    int*   tok_slot = (int*)  carve((size_t)T_TOK * 2 * sizeof(int));
    int*   slot_tok = (int*)  carve((size_t)NEXP * CAP * sizeof(int));
    float* slot_w   = (float*)carve((size_t)NEXP * CAP * sizeof(float));
    float* Gbuf     = (float*)carve((size_t)T_TOK * FDIM * sizeof(float));   // reused shared->routed
    float* Yr       = (float*)carve((size_t)NEXP * CAP * DIM * sizeof(float));

    // 1) init slot tables
    moe_init_kernel<<<(T_TOK * 2 + 255) / 256, 256, 0, stream>>>(tok_slot, slot_tok, slot_w);
    // 2) gating
    moe_gate_kernel<<<T_TOK, 256, 0, stream>>>(x, gate_w, bias, aff, top2);
    // 3) per-expert capacity selection (stable ranks)
    moe_select_kernel<<<dim3(T_TOK / 256, NEXP), 256, 0, stream>>>(aff, top2, slot_tok, slot_w, tok_slot);
    // 4) shared expert: G = swiglu(X @ W1), out = G @ W2
    moe_gemm1_swiglu_kernel<<<dim3(FDIM / BN, T_TOK / BM, 1), 256, 0, stream>>>(
        x, nullptr, sw1, 0, Gbuf, 0, T_TOK);
    moe_gemm2_kernel<<<dim3(DIM / BN, T_TOK / BM, 1), 256, 0, stream>>>(
        Gbuf, 0, sw2, 0, out, 0, T_TOK);
    // 5) routed experts (gathered rows via slot_tok), reuse Gbuf
    const int rby = (CAP + BM - 1) / BM;
    moe_gemm1_swiglu_kernel<<<dim3(FDIM / BN, rby, NEXP), 256, 0, stream>>>(
        x, slot_tok, rw1, (long)DIM * 2 * FDIM, Gbuf, (long)CAP * FDIM, CAP);
    moe_gemm2_kernel<<<dim3(DIM / BN, rby, NEXP), 256, 0, stream>>>(
        Gbuf, (long)CAP * FDIM, rw2, (long)FDIM * DIM, Yr, (long)CAP * DIM, CAP);
    // 6) weighted gather-add into output
    moe_gather_kernel<<<(int)(((size_t)T_TOK * DIM / 4) / 256), 256, 0, stream>>>(
        out, Yr, tok_slot, slot_w);
}